// Codebook_82678120448304
// MI455X (gfx1250) — compile-verified
//
#include <hip/hip_runtime.h>
#include <hip/hip_bf16.h>

// ---------------------------------------------------------------------------
// VQ codebook assignment, fused GEMM(+argmin) on CDNA5 WMMA f32 16x16x4.
//   z   : [4096 x 144] f32      (4*1024*144)
//   emb : [50257 x 144] f32
//   out : z_q [589824] f32  ++  indices [4096] (as f32)  ++  loss [1] f32
// ---------------------------------------------------------------------------

#define N_DIM       144
#define NUM_CODES   50257
#define N_ROWS      4096            // 4*1024
#define N_ROWTILES  (N_ROWS / 16)   // 256
#define P_PART      30
#define CODES_PER_PART ((NUM_CODES + P_PART - 1) / P_PART)   // 1676
#define TILES_PER_PART ((CODES_PER_PART + 15) / 16)          // 105
#define N_CHUNKS    (N_DIM / 4)     // 36 WMMA k-steps
#define ZT_STRIDE   148             // padded LDS row stride (banks coprime-ish)
#define ZQ_ELEMS    (N_ROWS * N_DIM)   // 589824

// workspace layout (in floats)
#define WS_NORMS 0                       // 50257 floats
#define WS_PARTV 50304                   // P_PART*N_ROWS = 122880 floats
#define WS_PARTI (50304 + 122880)        // 122880 ints
#define WS_ROWP  (50304 + 2 * 122880)    // 4096 floats

typedef __attribute__((ext_vector_type(2))) float v2f;
typedef __attribute__((ext_vector_type(8))) float v8f;

// ---------------------------------------------------------------------------
// Kernel 1: code-vector squared norms
// ---------------------------------------------------------------------------
__global__ void cb_norms_kernel(const float* __restrict__ emb,
                                float* __restrict__ norms) {
    int k = blockIdx.x * blockDim.x + threadIdx.x;
    if (k >= NUM_CODES) return;
    const float4* er = (const float4*)(emb + (size_t)k * N_DIM);
    float s = 0.f;
#pragma unroll
    for (int j = 0; j < N_DIM / 4; ++j) {
        float4 v = er[j];
        s += v.x * v.x + v.y * v.y + v.z * v.z + v.w * v.w;
    }
    norms[k] = s;
}

// ---------------------------------------------------------------------------
// Kernel 2: fused WMMA score + argmin over a code partition.
//   blockIdx.x = row tile (16 rows), blockIdx.y = code partition.
//   256 threads = 8 waves; each wave owns interleaved 16-code tiles.
// ---------------------------------------------------------------------------
__global__ __launch_bounds__(256)
void cb_argmin_kernel(const float* __restrict__ z,
                      const float* __restrict__ emb,
                      const float* __restrict__ norms,
                      float* __restrict__ partV,
                      int*   __restrict__ partI) {
    __shared__ float zt[16 * ZT_STRIDE];
    __shared__ float redV[8][16];
    __shared__ int   redI[8][16];

    const int rt   = blockIdx.x;
    const int p    = blockIdx.y;
    const int tid  = threadIdx.x;
    const int wave = tid >> 5;
    const int lane = tid & 31;

    // stage the 16x144 z tile into LDS (padded rows)
    for (int i = tid; i < 16 * N_DIM; i += 256) {
        int r = i / N_DIM, c = i - r * N_DIM;
        zt[r * ZT_STRIDE + c] = z[(size_t)rt * 16 * N_DIM + i];
    }
    __syncthreads();

    const float INF = __builtin_inff();
    const int   n   = lane & 15;        // N column of the 16x16 tile
    const int   kbo = (lane >> 4) * 2;  // K sub-offset per half-wave

    float minv[8];
    int   mini[8];
#pragma unroll
    for (int r = 0; r < 8; ++r) { minv[r] = INF; mini[r] = 0x7fffffff; }

    const float* zrow = &zt[n * ZT_STRIDE];   // A fragment: M = lane&15

    for (int t = wave; t < TILES_PER_PART; t += 8) {
        const int c0   = p * CODES_PER_PART + t * 16;
        const int code = c0 + n;
        const bool valid = (code < NUM_CODES);
        const float* erow = emb + (size_t)(valid ? code : 0) * N_DIM;

        // prefetch next tile's code row for this lane
        int cnext = c0 + 8 * 16 + n;
        if (t + 8 < TILES_PER_PART && cnext < NUM_CODES)
            __builtin_prefetch(emb + (size_t)cnext * N_DIM, 0, 1);

        v8f acc = {};
#pragma unroll
        for (int ch = 0; ch < N_CHUNKS; ++ch) {
            const int kb = ch * 4 + kbo;
            v2f a = *(const v2f*)(&zrow[kb]);     // LDS ds_load_b64
            v2f b = *(const v2f*)(&erow[kb]);     // global_load_b64 (L2-resident)
            acc = __builtin_amdgcn_wmma_f32_16x16x4_f32(
                false, a, false, b, (short)0, acc, false, false);
        }

        const float nv = valid ? norms[code] : INF;   // INF masks OOB codes
#pragma unroll
        for (int r = 0; r < 8; ++r) {
            float s = nv - 2.0f * acc[r];             // ||e||^2 - 2 z.e
            if (s < minv[r] || (s == minv[r] && code < mini[r])) {
                minv[r] = s; mini[r] = code;
            }
        }
    }

    // reduce across the 16 N-lanes (xor stays inside each 16-lane group)
#pragma unroll
    for (int m = 1; m < 16; m <<= 1) {
#pragma unroll
        for (int r = 0; r < 8; ++r) {
            float ov = __shfl_xor(minv[r], m, 32);
            int   oi = __shfl_xor(mini[r], m, 32);
            if (ov < minv[r] || (ov == minv[r] && oi < mini[r])) {
                minv[r] = ov; mini[r] = oi;
            }
        }
    }
    // C/D layout: row M = r + 8*(lane>>4); lane 0 publishes rows 0..7,
    // lane 16 publishes rows 8..15.
    if (lane == 0 || lane == 16) {
        int base = (lane >> 4) * 8;
#pragma unroll
        for (int r = 0; r < 8; ++r) {
            redV[wave][base + r] = minv[r];
            redI[wave][base + r] = mini[r];
        }
    }
    __syncthreads();

    // cross-wave reduce, one thread per row
    if (tid < 16) {
        float bv = redV[0][tid];
        int   bi = redI[0][tid];
#pragma unroll
        for (int w = 1; w < 8; ++w) {
            float v = redV[w][tid];
            int   i = redI[w][tid];
            if (v < bv || (v == bv && i < bi)) { bv = v; bi = i; }
        }
        const int row = rt * 16 + tid;
        partV[(size_t)p * N_ROWS + row] = bv;
        partI[(size_t)p * N_ROWS + row] = bi;
    }
}

// ---------------------------------------------------------------------------
// Kernel 3: per-row partition reduce, gather z_q, per-row loss partial.
//   One block (256 thr) per row.
// ---------------------------------------------------------------------------
__global__ __launch_bounds__(256)
void cb_gather_kernel(const float* __restrict__ z,
                      const float* __restrict__ emb,
                      const float* __restrict__ partV,
                      const int*   __restrict__ partI,
                      float* __restrict__ out,
                      float* __restrict__ rowP) {
    __shared__ int   sIdx;
    __shared__ float sred[256];

    const int row = blockIdx.x;
    const int tid = threadIdx.x;

    if (tid == 0) {
        float bv = __builtin_inff();
        int   bi = 0;
        for (int p = 0; p < P_PART; ++p) {
            float v = partV[(size_t)p * N_ROWS + row];
            int   i = partI[(size_t)p * N_ROWS + row];
            if (v < bv || (v == bv && i < bi)) { bv = v; bi = i; }
        }
        sIdx = bi;
    }
    __syncthreads();
    const int idx = sIdx;

    float a = 0.f;
    if (tid < N_DIM) {
        float zq = emb[(size_t)idx * N_DIM + tid];
        out[(size_t)row * N_DIM + tid] = zq;
        float d = zq - z[(size_t)row * N_DIM + tid];
        a = d * d;
    }
    sred[tid] = a;
    __syncthreads();
#pragma unroll
    for (int s = 128; s > 0; s >>= 1) {
        if (tid < s) sred[tid] += sred[tid + s];
        __syncthreads();
    }
    if (tid == 0) {
        rowP[row] = sred[0];
        out[ZQ_ELEMS + row] = (float)idx;   // indices emitted after z_q
    }
}

// ---------------------------------------------------------------------------
// Kernel 4: final loss mean
// ---------------------------------------------------------------------------
__global__ __launch_bounds__(256)
void cb_loss_kernel(const float* __restrict__ rowP, float* __restrict__ out) {
    __shared__ float sred[256];
    const int tid = threadIdx.x;
    float a = 0.f;
    for (int i = tid; i < N_ROWS; i += 256) a += rowP[i];
    sred[tid] = a;
    __syncthreads();
#pragma unroll
    for (int s = 128; s > 0; s >>= 1) {
        if (tid < s) sred[tid] += sred[tid + s];
        __syncthreads();
    }
    if (tid == 0) out[ZQ_ELEMS + N_ROWS] = sred[0] / (float)ZQ_ELEMS;
}

// ---------------------------------------------------------------------------
extern "C" void kernel_launch(void* const* d_in, const int* in_sizes, int n_in,
                              void* d_out, int out_size, void* d_ws, size_t ws_size,
                              hipStream_t stream) {
    const float* z   = (const float*)d_in[0];
    const float* emb = (const float*)d_in[1];
    // d_in[2] = pkeep (== 0 here; random-replacement branch disabled)

    float* out   = (float*)d_out;
    float* ws    = (float*)d_ws;
    float* norms = ws + WS_NORMS;
    float* partV = ws + WS_PARTV;
    int*   partI = (int*)(ws + WS_PARTI);
    float* rowP  = ws + WS_ROWP;

    cb_norms_kernel<<<(NUM_CODES + 255) / 256, 256, 0, stream>>>(emb, norms);

    dim3 grid(N_ROWTILES, P_PART);
    cb_argmin_kernel<<<grid, 256, 0, stream>>>(z, emb, norms, partV, partI);

    cb_gather_kernel<<<N_ROWS, 256, 0, stream>>>(z, emb, partV, partI, out, rowP);

    cb_loss_kernel<<<1, 256, 0, stream>>>(rowP, out);
}